// ProbAttention_44315472560730
// MI455X (gfx1250) — compile-verified
//
#include <hip/hip_runtime.h>
#include <hip/hip_bf16.h>
#include <math.h>

typedef __attribute__((ext_vector_type(2))) float v2f;
typedef __attribute__((ext_vector_type(8))) float v8f;

#define B_ 4
#define T_ 512
#define N_ 4
#define H_ 8
#define D_ 64
#define E_ 256   // N_*D_
#define U_ 35
#define UP_ 48   // padded to 3 x 16-row tiles
#define BH_ 32   // B_*H_
#define NEG_INF (-3.402823466e+38f)

// flat index into the original (b,t,n,h,d) arrays, addressed as merged (b,h,t,e)
__device__ __forceinline__ size_t gidx(int b, int t, int e, int h) {
  return (size_t)b * (T_ * N_ * H_ * D_) + (size_t)t * (N_ * H_ * D_) +
         (size_t)(e >> 6) * (H_ * D_) + (size_t)h * D_ + (size_t)(e & 63);
}

// ---------------- Kernel 1: sparse sampled metric M[b,h,t] ----------------
__global__ void k_compute_M(const float* __restrict__ q, const float* __restrict__ k,
                            const int* __restrict__ idx, float* __restrict__ M) {
  int wave = (int)((blockIdx.x * blockDim.x + threadIdx.x) >> 5);
  int lane = threadIdx.x & 31;
  if (wave >= B_ * H_ * T_) return;
  int t = wave % T_;
  int bh = wave / T_;
  int h = bh % H_;
  int b = bh / H_;

  float qr[8];
#pragma unroll
  for (int j = 0; j < 8; ++j) {
    int e = lane + 32 * j;
    qr[j] = q[gidx(b, t, e, h)];
  }
  float mx = NEG_INF, sm = 0.f;
  for (int s = 0; s < U_; ++s) {
    int ts = idx[t * U_ + s];
    float p = 0.f;
#pragma unroll
    for (int j = 0; j < 8; ++j) {
      int e = lane + 32 * j;
      p += qr[j] * k[gidx(b, ts, e, h)];
    }
#pragma unroll
    for (int off = 16; off > 0; off >>= 1) p += __shfl_xor(p, off, 32);
    mx = fmaxf(mx, p);
    sm += p;
  }
  if (lane == 0) M[bh * T_ + t] = mx - sm * (1.0f / (float)T_);
}

// ---------------- Kernel 2: top-35 per (b,h), padded to 48 ----------------
__global__ void k_topk(const float* __restrict__ M, int* __restrict__ Mtop) {
  __shared__ float vals[T_];
  __shared__ float rv[T_];
  __shared__ int ri[T_];
  int bh = blockIdx.x;
  int tid = threadIdx.x;  // 512 threads
  vals[tid] = M[bh * T_ + tid];
  __syncthreads();
  int last = 0;
  for (int it = 0; it < U_; ++it) {
    rv[tid] = vals[tid];
    ri[tid] = tid;
    __syncthreads();
    for (int off = T_ / 2; off > 0; off >>= 1) {
      if (tid < off) {
        float a = rv[tid], c = rv[tid + off];
        if (c > a || (c == a && ri[tid + off] < ri[tid])) {
          rv[tid] = c;
          ri[tid] = ri[tid + off];
        }
      }
      __syncthreads();
    }
    int w = ri[0];
    if (tid == 0) Mtop[bh * UP_ + it] = w;
    if (tid == w) vals[tid] = NEG_INF;
    last = w;
    __syncthreads();
  }
  if (tid == 0) {
    for (int it = U_; it < UP_; ++it) Mtop[bh * UP_ + it] = last;  // pad rows
  }
}

// ---------------- Kernel 3: column mean of V ----------------
__global__ void k_meanv(const float* __restrict__ v, float* __restrict__ meanV) {
  int bh = blockIdx.x;
  int e = threadIdx.x;  // 256 threads
  int h = bh & 7, b = bh >> 3;
  float s = 0.f;
  for (int t = 0; t < T_; ++t) s += v[gidx(b, t, e, h)];
  meanV[bh * E_ + e] = s * (1.0f / (float)T_);
}

// ---------------- Kernel 4: broadcast mean into output ----------------
__global__ void k_fill(const float* __restrict__ meanV, float4* __restrict__ out) {
  int i = blockIdx.x * blockDim.x + threadIdx.x;  // out_size/4 = 1048576
  if (i >= (B_ * T_ * N_ * H_ * D_) / 4) return;
  int base = i * 4;
  int dd = base & 63;
  int h = (base >> 6) & 7;
  int n = (base >> 9) & 3;
  int b = base >> 20;
  const float4* mv = (const float4*)meanV;
  out[i] = mv[(((b * H_ + h) * E_) + n * D_ + dd) >> 2];
}

// ---------------- Kernel 5: scores = Q_top @ K^T * (1/8) via WMMA f32 ----------------
__global__ void __launch_bounds__(256) k_scores(const float* __restrict__ q,
                                                const float* __restrict__ kk,
                                                const int* __restrict__ Mtop,
                                                float* __restrict__ scores) {
  __shared__ int sTop[UP_];
  int bh = blockIdx.x;
  int b = bh >> 3, h = bh & 7;
  int tid = threadIdx.x;
  if (tid < UP_) sTop[tid] = Mtop[bh * UP_ + tid];
  __syncthreads();

  int wave = tid >> 5, lane = tid & 31;
  int lm = lane & 15;
  int khalf = (lane >> 4) << 1;  // 0 or 2
  int ncol0 = wave * 64;         // 8 waves x 64 cols = 512

  v8f zero = {0.f, 0.f, 0.f, 0.f, 0.f, 0.f, 0.f, 0.f};
  v8f acc[3][4];
#pragma unroll
  for (int i = 0; i < 3; ++i)
#pragma unroll
    for (int j = 0; j < 4; ++j) acc[i][j] = zero;

  for (int ks = 0; ks < E_ / 4; ++ks) {
    int kb = ks * 4 + khalf;  // even element index, pair (kb, kb+1) contiguous
    v2f afr[3];
#pragma unroll
    for (int i = 0; i < 3; ++i) {
      int ta = sTop[i * 16 + lm];
      afr[i] = *(const v2f*)(q + gidx(b, ta, kb, h));
    }
    v2f bfr[4];
#pragma unroll
    for (int j = 0; j < 4; ++j) {
      int tb = ncol0 + j * 16 + lm;
      bfr[j] = *(const v2f*)(kk + gidx(b, tb, kb, h));
    }
#pragma unroll
    for (int i = 0; i < 3; ++i)
#pragma unroll
      for (int j = 0; j < 4; ++j)
        acc[i][j] = __builtin_amdgcn_wmma_f32_16x16x4_f32(false, afr[i], false, bfr[j],
                                                          (short)0, acc[i][j], false, false);
  }

  // write scaled scores: row m = i*16 + r + 8*(lane>=16), col = ncol0 + j*16 + lm
#pragma unroll
  for (int i = 0; i < 3; ++i) {
    int mbase = i * 16 + ((lane >> 4) << 3);
#pragma unroll
    for (int j = 0; j < 4; ++j) {
      int col = ncol0 + j * 16 + lm;
#pragma unroll
      for (int r = 0; r < 8; ++r)
        scores[((size_t)bh * UP_ + (mbase + r)) * T_ + col] = acc[i][j][r] * 0.125f;
    }
  }
}

// ---------------- Kernel 6: row softmax over 512 (one wave per row) ----------------
__global__ void k_softmax(float* __restrict__ scores) {
  int wave = (int)((blockIdx.x * blockDim.x + threadIdx.x) >> 5);
  int lane = threadIdx.x & 31;
  if (wave >= BH_ * UP_) return;
  float* row = scores + (size_t)wave * T_;
  float vals[16];
  float mx = NEG_INF;
#pragma unroll
  for (int j = 0; j < 16; ++j) {
    vals[j] = row[lane + 32 * j];
    mx = fmaxf(mx, vals[j]);
  }
#pragma unroll
  for (int off = 16; off > 0; off >>= 1) mx = fmaxf(mx, __shfl_xor(mx, off, 32));
  float s = 0.f;
#pragma unroll
  for (int j = 0; j < 16; ++j) {
    vals[j] = __expf(vals[j] - mx);
    s += vals[j];
  }
#pragma unroll
  for (int off = 16; off > 0; off >>= 1) s += __shfl_xor(s, off, 32);
  float inv = 1.0f / s;
#pragma unroll
  for (int j = 0; j < 16; ++j) row[lane + 32 * j] = vals[j] * inv;
}

// ---------------- Kernel 7: update = attn @ V via WMMA, scatter into out ----------------
__global__ void __launch_bounds__(256) k_update(const float* __restrict__ attn,
                                                const float* __restrict__ v,
                                                const int* __restrict__ Mtop,
                                                float* __restrict__ out) {
  __shared__ int sTop[UP_];
  int bh = blockIdx.x;
  int b = bh >> 3, h = bh & 7;
  int tid = threadIdx.x;
  if (tid < UP_) sTop[tid] = Mtop[bh * UP_ + tid];
  __syncthreads();

  int wave = tid >> 5, lane = tid & 31;
  int lm = lane & 15;
  int khalf = (lane >> 4) << 1;
  int ncol0 = wave * 32;  // 8 waves x 32 cols = 256 (e-dim)

  v8f zero = {0.f, 0.f, 0.f, 0.f, 0.f, 0.f, 0.f, 0.f};
  v8f acc[3][2];
#pragma unroll
  for (int i = 0; i < 3; ++i)
#pragma unroll
    for (int j = 0; j < 2; ++j) acc[i][j] = zero;

  const float* arow = attn + (size_t)bh * UP_ * T_;
  for (int ks = 0; ks < T_ / 4; ++ks) {
    int kb = ks * 4 + khalf;
    v2f afr[3];
#pragma unroll
    for (int i = 0; i < 3; ++i)
      afr[i] = *(const v2f*)(arow + (size_t)(i * 16 + lm) * T_ + kb);
    v2f bfr[2];
#pragma unroll
    for (int j = 0; j < 2; ++j) {
      int e = ncol0 + j * 16 + lm;
      bfr[j].x = v[gidx(b, kb, e, h)];
      bfr[j].y = v[gidx(b, kb + 1, e, h)];
    }
#pragma unroll
    for (int i = 0; i < 3; ++i)
#pragma unroll
      for (int j = 0; j < 2; ++j)
        acc[i][j] = __builtin_amdgcn_wmma_f32_16x16x4_f32(false, afr[i], false, bfr[j],
                                                          (short)0, acc[i][j], false, false);
  }

  // scatter valid rows (m < 35) into output at t = sTop[m]
#pragma unroll
  for (int i = 0; i < 3; ++i) {
    int mbase = i * 16 + ((lane >> 4) << 3);
#pragma unroll
    for (int r = 0; r < 8; ++r) {
      int m = mbase + r;
      if (m < U_) {
        int tOut = sTop[m];
#pragma unroll
        for (int j = 0; j < 2; ++j) {
          int e = ncol0 + j * 16 + lm;
          out[gidx(b, tOut, e, h)] = acc[i][j][r];
        }
      }
    }
  }
}

extern "C" void kernel_launch(void* const* d_in, const int* in_sizes, int n_in,
                              void* d_out, int out_size, void* d_ws, size_t ws_size,
                              hipStream_t stream) {
  const float* q = (const float*)d_in[0];
  const float* k = (const float*)d_in[1];
  const float* v = (const float*)d_in[2];
  const int* idx = (const int*)d_in[3];
  float* out = (float*)d_out;

  char* ws = (char*)d_ws;
  float* M = (float*)ws;                  // 16384 floats  (64 KB)
  int* Mtop = (int*)(ws + 65536);         // 1536 ints     (6 KB)
  float* meanV = (float*)(ws + 71680);    // 8192 floats   (32 KB)
  float* scores = (float*)(ws + 104448);  // 786432 floats (3 MB)

  k_compute_M<<<(B_ * H_ * T_) / 8, 256, 0, stream>>>(q, k, idx, M);
  k_topk<<<BH_, 512, 0, stream>>>(M, Mtop);
  k_meanv<<<BH_, 256, 0, stream>>>(v, meanV);
  k_fill<<<(B_ * T_ * N_ * H_ * D_) / 4 / 256, 256, 0, stream>>>(meanV, (float4*)out);
  k_scores<<<BH_, 256, 0, stream>>>(q, k, Mtop, scores);
  k_softmax<<<(BH_ * UP_) / 8, 256, 0, stream>>>(scores);
  k_update<<<BH_, 256, 0, stream>>>(scores, v, Mtop, out);
}